// DiffAugment_48498770706988
// MI455X (gfx1250) — compile-verified
//
#include <hip/hip_runtime.h>

// DiffAugment: flip -> brightness -> contrast -> translation -> cutout
// Single-pass gather + affine + predicated-zero. Memory-bound:
// ~100.7 MB traffic => ~4.3 us floor at 23.3 TB/s on MI455X.
// Per-batch params forced into SGPRs (readfirstlane) => uniform scalar branches.

#define BB 64
#define CC 3
#define HH 256
#define WW 256

typedef float vfloat4 __attribute__((ext_vector_type(4)));

__device__ __forceinline__ float rflf(float v) {
    return __int_as_float(__builtin_amdgcn_readfirstlane(__float_as_int(v)));
}
__device__ __forceinline__ int rfli(int v) {
    return __builtin_amdgcn_readfirstlane(v);
}

__global__ __launch_bounds__(256) void
diffaug_kernel(const float* __restrict__ x,
               const float* __restrict__ p_arr,
               const float* __restrict__ flip_u,
               const float* __restrict__ bright_n,
               const float* __restrict__ bright_u,
               const float* __restrict__ contrast_n,
               const float* __restrict__ contrast_u,
               const int*   __restrict__ trans_h,
               const int*   __restrict__ trans_w,
               const float* __restrict__ trans_u,
               const int*   __restrict__ cut_ox,
               const int*   __restrict__ cut_oy,
               const float* __restrict__ cut_u,
               float* __restrict__ out)
{
    const int tid      = threadIdx.x;
    const int rowInBlk = tid >> 6;          // 0..3 (4 rows per 256-thread block)
    const int lane64   = tid & 63;          // position within row (4 px each)
    const int row      = blockIdx.x * 4 + rowInBlk;   // [0, B*C*H)
    const int h        = row & (HH - 1);
    const int bc       = row >> 8;          // b*C + c  (H==256 rows per plane)
    const int b        = bc / CC;

    const float* __restrict__ plane  = x   + (size_t)bc * (HH * WW);
    float*       __restrict__ oplane = out + (size_t)bc * (HH * WW);

    // ---- per-batch params -> SGPRs via readfirstlane (block-uniform) ----
    const float pv    = p_arr[0];
    const bool  flip  = rflf(flip_u[b])     < 0.5f * pv;
    const bool  doBr  = rflf(bright_u[b])   < pv;
    const bool  doCo  = rflf(contrast_u[b]) < pv;
    const bool  doTr  = rflf(trans_u[b])    < pv;
    const bool  doCut = rflf(cut_u[b])      < pv;
    const float bAdd  = doBr ? rflf(bright_n[b]) * 0.2f : 0.0f;
    const float cMul  = doCo ? exp2f(rflf(contrast_n[b]) * 0.5f) : 1.0f;
    const int   th    = rfli(trans_h[b]);
    const int   tw    = rfli(trans_w[b]);
    const int   ox    = rfli(cut_ox[b]);
    const int   oy    = rfli(cut_oy[b]);

    // cutout rectangle (clamped scatter == clamped interval)
    const int cutLoH = max(ox - 64, 0);
    const int cutHiH = min(ox + 63, HH - 1);
    const int cutLoW = max(oy - 64, 0);
    const int cutHiW = min(oy + 63, WW - 1);

    const int w0 = lane64 * 4;
    float v0, v1, v2, v3;

    // ---- resolve source row (translation w/ torch pad-by-1 quirk) ----
    int  srcRow   = h;
    bool rowValid = true;
    if (doTr) {
        int g = h + th + 1;                          // index into H+2 padded dim
        g = g < 0 ? 0 : (g > HH + 1 ? HH + 1 : g);   // clamp to [0, H+1]
        srcRow   = g - 1;                            // -1 or H => zero pad row
        rowValid = (srcRow >= 0) && (srcRow < HH);   // uniform per (block row)
    }

    if (!rowValid) {
        v0 = v1 = v2 = v3 = 0.0f;                    // pad zeros skip bAdd/cMul
    } else {
        const float* __restrict__ rowp = plane + srcRow * WW;
        if (!doTr) {
            if (!flip) {
                const vfloat4 t = *reinterpret_cast<const vfloat4*>(rowp + w0);
                v0 = t.x; v1 = t.y; v2 = t.z; v3 = t.w;
            } else {
                // mirrored quad is still a 16B-aligned b128 load; reverse in regs
                const vfloat4 t = *reinterpret_cast<const vfloat4*>(rowp + (WW - 4 - w0));
                v0 = t.w; v1 = t.z; v2 = t.y; v3 = t.x;
            }
        } else {
            // warm source row in GL2 before strided gather (gfx1250 prefetch)
            if (lane64 == 0) __builtin_prefetch(rowp, 0, 1);
            // w' = (w + tw) mod (W-1); one mod, then incremental wrap
            int sw = (w0 + tw + 255) % 255;          // tw in [-16,16] => arg >= 239
            float fv[4];
            #pragma unroll
            for (int k = 0; k < 4; ++k) {
                const int swe = flip ? (WW - 1 - sw) : sw;
                fv[k] = rowp[swe];
                sw = (sw == 254) ? 0 : sw + 1;
            }
            v0 = fv[0]; v1 = fv[1]; v2 = fv[2]; v3 = fv[3];
        }
        v0 = (v0 + bAdd) * cMul;
        v1 = (v1 + bAdd) * cMul;
        v2 = (v2 + bAdd) * cMul;
        v3 = (v3 + bAdd) * cMul;
    }

    // ---- cutout (applied after translation select) ----
    if (doCut && h >= cutLoH && h <= cutHiH) {
        if (w0 + 0 >= cutLoW && w0 + 0 <= cutHiW) v0 = 0.0f;
        if (w0 + 1 >= cutLoW && w0 + 1 <= cutHiW) v1 = 0.0f;
        if (w0 + 2 >= cutLoW && w0 + 2 <= cutHiW) v2 = 0.0f;
        if (w0 + 3 >= cutLoW && w0 + 3 <= cutHiW) v3 = 0.0f;
    }

    vfloat4 o; o.x = v0; o.y = v1; o.z = v2; o.w = v3;
    // streaming (non-temporal) b128 store: write-once output, keep input in L2
    __builtin_nontemporal_store(o, reinterpret_cast<vfloat4*>(oplane + h * WW + w0));
}

extern "C" void kernel_launch(void* const* d_in, const int* in_sizes, int n_in,
                              void* d_out, int out_size, void* d_ws, size_t ws_size,
                              hipStream_t stream) {
    (void)in_sizes; (void)n_in; (void)out_size; (void)d_ws; (void)ws_size;
    const float* x          = (const float*)d_in[0];
    const float* p          = (const float*)d_in[1];
    const float* flip_u     = (const float*)d_in[2];
    const float* bright_n   = (const float*)d_in[3];
    const float* bright_u   = (const float*)d_in[4];
    const float* contrast_n = (const float*)d_in[5];
    const float* contrast_u = (const float*)d_in[6];
    const int*   trans_h    = (const int*)d_in[7];
    const int*   trans_w    = (const int*)d_in[8];
    const float* trans_u    = (const float*)d_in[9];
    const int*   cut_ox     = (const int*)d_in[10];
    const int*   cut_oy     = (const int*)d_in[11];
    const float* cut_u      = (const float*)d_in[12];
    float* out = (float*)d_out;

    const int totalRows = BB * CC * HH;           // 49152 rows of 256 px
    dim3 grid(totalRows / 4);                     // 4 rows per block
    dim3 block(256);                              // 8 wave32 waves per block
    diffaug_kernel<<<grid, block, 0, stream>>>(x, p, flip_u, bright_n, bright_u,
                                               contrast_n, contrast_u,
                                               trans_h, trans_w, trans_u,
                                               cut_ox, cut_oy, cut_u, out);
}